// MaterialDecoder_20796231647234
// MI455X (gfx1250) — compile-verified
//
#include <hip/hip_runtime.h>

typedef __attribute__((ext_vector_type(2))) float v2f;
typedef __attribute__((ext_vector_type(8))) float v8f;

#define N_ROWS   1000000
#define ELE_DIM  32
#define MAT_FEAT 83

// One wave (32 lanes) computes a 16-row x 16-feature tile of
//   y = inputs @ W^T + b  via 8 chained V_WMMA_F32_16X16X4_F32 ops (K=32).
// Block = 192 threads = 6 waves = 6 feature tiles covering 83 features.
// Grid.x = N/16 row tiles.
//
// Cache policy (MI455X, 23.3 TB/s HBM, 192MB L2; 464MB traffic/call):
//   - inputs: read-once stream  -> non-temporal loads (evict early)
//   - W/bias: 10.6KB, reused by every block -> regular temporal loads
//   - output: 336MB write-once stream -> non-temporal stores (no L2 pollution)
__global__ __launch_bounds__(192, 1)
void material_decoder_wmma(const float* __restrict__ inp,   // [N,32]
                           const float* __restrict__ Wm,    // [83,32]
                           const float* __restrict__ bias,  // [83]
                           float* __restrict__ out)         // [N*83] x, then [N] mask
{
    const unsigned lane = threadIdx.x & 31u;
    const unsigned wave = threadIdx.x >> 5;      // feature tile 0..5
    const unsigned m    = lane & 15u;            // row (A) / col (B,C,D) within tile
    const unsigned hi   = lane >> 4;             // lane-half selector
    const unsigned kb   = hi * 2u;               // K offset inside each 4-chunk

    const long long row0 = (long long)blockIdx.x * 16;
    const long long row  = row0 + m;             // global input row for A fragment

    const int fcol = (int)(wave * 16u + m);                   // global feature col
    const int fcl  = fcol < MAT_FEAT ? fcol : (MAT_FEAT - 1); // clamped for loads

    const float* arow = inp + row * ELE_DIM;
    const float* wrow = Wm + (long long)fcl * ELE_DIM;

    // ---- load A and B fragments for all 8 K-steps; track per-lane nonzero ----
    v2f a[8], bb[8];
    unsigned nzacc = 0u;
    #pragma unroll
    for (int ks = 0; ks < 8; ++ks) {
        a[ks]  = __builtin_nontemporal_load((const v2f*)(arow + ks * 4 + kb));
        bb[ks] = *(const v2f*)(wrow + ks * 4 + kb);
        nzacc |= (unsigned)(a[ks].x != 0.0f) | (unsigned)(a[ks].y != 0.0f);
    }

    // ---- 16x16x32 f32 GEMM tile: chain of 8 WMMAs (EXEC all ones, no branches) ----
    v8f c = {};
    #pragma unroll
    for (int ks = 0; ks < 8; ++ks) {
        c = __builtin_amdgcn_wmma_f32_16x16x4_f32(
                /*neg_a=*/false, a[ks], /*neg_b=*/false, bb[ks],
                /*c_mod=*/(short)0, c, /*reuse_a=*/false, /*reuse_b=*/false);
    }

    // ---- row-nonzero mask: lanes m and m+16 jointly hold row m's 32 values ----
    unsigned bal = __builtin_amdgcn_ballot_w32(nzacc != 0u);
    unsigned m16 = (bal | (bal >> 16)) & 0xFFFFu;   // bit r = row r has any nonzero

    // ---- epilogue: bias + masked sigmoid, coalesced non-temporal stores ----
    // C/D layout: VGPR i, lane L -> row M = i + 8*(L>=16), col N = L%16.
    const float bv = bias[fcl];
    #pragma unroll
    for (int i = 0; i < 8; ++i) {
        const unsigned rm  = (unsigned)i + hi * 8u;
        const bool     rnz = ((m16 >> rm) & 1u) != 0u;
        const float    y   = c[i] + bv;
        const float    x   = rnz ? (1.0f / (1.0f + __expf(-y))) : 0.5f;
        if (fcol < MAT_FEAT)
            __builtin_nontemporal_store(x, out + (row0 + (long long)rm) * MAT_FEAT + fcol);
    }

    // ---- mask output: one store per row, done by wave 0 lanes 0..15 ----
    if (wave == 0u && hi == 0u) {
        const float mv = ((m16 >> m) & 1u) ? 1.0f : 0.0f;
        __builtin_nontemporal_store(mv, out + (long long)N_ROWS * MAT_FEAT + row);
    }
}

extern "C" void kernel_launch(void* const* d_in, const int* in_sizes, int n_in,
                              void* d_out, int out_size, void* d_ws, size_t ws_size,
                              hipStream_t stream) {
    (void)in_sizes; (void)n_in; (void)d_ws; (void)ws_size; (void)out_size;
    const float* inp  = (const float*)d_in[0];  // [N,32]
    const float* Wm   = (const float*)d_in[1];  // [83,32]
    const float* bias = (const float*)d_in[2];  // [83]
    float* out = (float*)d_out;                 // [N*83 + N]

    const int row_tiles = N_ROWS / 16;          // 62500
    material_decoder_wmma<<<dim3(row_tiles), dim3(192), 0, stream>>>(inp, Wm, bias, out);
}